// FocalLoss_twologits_40999757807958
// MI455X (gfx1250) — compile-verified
//
#include <hip/hip_runtime.h>

typedef __attribute__((ext_vector_type(2))) float v2f;
typedef __attribute__((ext_vector_type(4))) float v4f;
typedef __attribute__((ext_vector_type(8))) float v8f;
typedef __attribute__((ext_vector_type(4))) int   v4i;

#define NBLOCKS 2048
#define NTHREADS 256

// Full-wave (32 lane) sum using V_WMMA_F32_16X16X4_F32 with B == ones.
// A layout (16x4 f32): A[m][0]=lane m .x, A[m][1]=lane m .y,
//                      A[m][2]=lane m+16 .x, A[m][3]=lane m+16 .y
// With A.x = per-lane value, A.y = 0, B = 1:  D[m][n] = x[m] + x[m+16].
// Lane L<16 holds D rows 0..7 in c[0..7]; lanes >=16 hold rows 8..15.
// Sum of own 8 c-regs + ds_swizzle SWAPX16 => full 32-lane sum on every lane.
// EXEC must be all 1s at the call site (callers guarantee full blocks).
__device__ __forceinline__ float wave_sum32(float x) {
    v2f a; a[0] = x;    a[1] = 0.0f;
    v2f b; b[0] = 1.0f; b[1] = 1.0f;
    v8f c = {};
    c = __builtin_amdgcn_wmma_f32_16x16x4_f32(
        /*neg_a=*/false, a, /*neg_b=*/false, b,
        /*c_mod=*/(short)0, c, /*reuse_a=*/false, /*reuse_b=*/false);
    float s = ((c[0] + c[1]) + (c[2] + c[3])) + ((c[4] + c[5]) + (c[6] + c[7]));
    // group-of-32 swizzle, xor_mask=0x10 (SWAPX16): 0x10<<10 | 0x1f = 0x401F
    int sw = __builtin_amdgcn_ds_swizzle(__float_as_int(s), 0x401F);
    return s + __int_as_float(sw);
}

// loss = alpha_t * sigmoid(d)^2 * softplus(d),  d = l_other - l_target.
// One exp, one log, one fast rcp per element.
__device__ __forceinline__ float focal_elem(float l0, float l1, int t) {
    bool  one = (t != 0);
    float dd  = l0 - l1;
    float d   = one ? dd : -dd;                   // l_other - l_target
    float e   = __expf(-fabsf(d));                // exp(-|d|) in (0,1]
    float r   = 1.0f + e;
    float u   = __builtin_amdgcn_rcpf(r);         // fast 1/(1+e)
    float sig = (d >= 0.0f) ? u : e * u;          // sigmoid(d) = 1 - p_t
    float sp  = __logf(r) + fmaxf(d, 0.0f);       // softplus(d), stable
    float alpha = one ? 0.25f : 0.75f;
    return alpha * (sig * sig) * sp;
}

__global__ __launch_bounds__(NTHREADS)
void focal_partial_kernel(const v4f* __restrict__ lg4,
                          const v4i* __restrict__ tg4,
                          const float* __restrict__ lg_flat,
                          const int*   __restrict__ tg_flat,
                          float* __restrict__ partials,
                          int noct, int n) {
    __shared__ float smem[NTHREADS / 32];

    float acc0 = 0.0f, acc1 = 0.0f;
    int stride = gridDim.x * blockDim.x;
    for (int j = blockIdx.x * blockDim.x + threadIdx.x; j < noct; j += stride) {
        // 8 elements/iter: 4x b128 logit + 2x b128 target streaming loads
        v4f p0 = __builtin_nontemporal_load(lg4 + 4 * j + 0);
        v4f p1 = __builtin_nontemporal_load(lg4 + 4 * j + 1);
        v4f p2 = __builtin_nontemporal_load(lg4 + 4 * j + 2);
        v4f p3 = __builtin_nontemporal_load(lg4 + 4 * j + 3);
        v4i t0 = __builtin_nontemporal_load(tg4 + 2 * j + 0);
        v4i t1 = __builtin_nontemporal_load(tg4 + 2 * j + 1);
        acc0 += focal_elem(p0[0], p0[1], t0[0]);
        acc1 += focal_elem(p0[2], p0[3], t0[1]);
        acc0 += focal_elem(p1[0], p1[1], t0[2]);
        acc1 += focal_elem(p1[2], p1[3], t0[3]);
        acc0 += focal_elem(p2[0], p2[1], t1[0]);
        acc1 += focal_elem(p2[2], p2[3], t1[1]);
        acc0 += focal_elem(p3[0], p3[1], t1[2]);
        acc1 += focal_elem(p3[2], p3[3], t1[3]);
    }
    float acc = acc0 + acc1;

    // scalar tail (n not multiple of 8); divergence is fine here --
    // the WMMA reduction below is executed unconditionally by all lanes
    int tail = noct * 8;
    int rem  = n - tail;
    if (blockIdx.x == 0 && (int)threadIdx.x < rem) {
        int i = tail + (int)threadIdx.x;
        acc += focal_elem(lg_flat[2 * i], lg_flat[2 * i + 1], tg_flat[i]);
    }

    float ws = wave_sum32(acc);               // EXEC all-1s here
    int lane = threadIdx.x & 31;
    int wid  = threadIdx.x >> 5;
    if (lane == 0) smem[wid] = ws;
    __syncthreads();
    if (threadIdx.x == 0) {
        float bsum = 0.0f;
        #pragma unroll
        for (int w = 0; w < NTHREADS / 32; ++w) bsum += smem[w];
        partials[blockIdx.x] = bsum;          // overwrite: no cross-call state
    }
}

__global__ __launch_bounds__(NTHREADS)
void focal_final_kernel(const float* __restrict__ partials, int npart,
                        float* __restrict__ out, float inv_n) {
    __shared__ float smem[NTHREADS / 32];
    float acc = 0.0f;
    for (int i = threadIdx.x; i < npart; i += blockDim.x) acc += partials[i];
    float ws = wave_sum32(acc);
    int lane = threadIdx.x & 31;
    int wid  = threadIdx.x >> 5;
    if (lane == 0) smem[wid] = ws;
    __syncthreads();
    if (threadIdx.x == 0) {
        float bsum = 0.0f;
        #pragma unroll
        for (int w = 0; w < NTHREADS / 32; ++w) bsum += smem[w];
        out[0] = bsum * inv_n;
    }
}

extern "C" void kernel_launch(void* const* d_in, const int* in_sizes, int n_in,
                              void* d_out, int out_size, void* d_ws, size_t ws_size,
                              hipStream_t stream) {
    (void)n_in; (void)out_size; (void)ws_size;
    const float* logits = (const float*)d_in[0];   // [N,2] f32
    const int*   target = (const int*)d_in[1];     // [N]   i32
    float* out      = (float*)d_out;
    float* partials = (float*)d_ws;                // NBLOCKS floats

    int n    = in_sizes[1];
    int noct = n >> 3;

    focal_partial_kernel<<<NBLOCKS, NTHREADS, 0, stream>>>(
        (const v4f*)logits, (const v4i*)target, logits, target,
        partials, noct, n);
    focal_final_kernel<<<1, NTHREADS, 0, stream>>>(
        partials, NBLOCKS, out, 1.0f / (float)n);
}